// SHEmbed_69406671503654
// MI455X (gfx1250) — compile-verified
//
#include <hip/hip_runtime.h>
#include <stdint.h>
#include <math.h>

// SH embed, L=4 (25 coefs), RES=512, f32.
// Wave-cooperative async gather of each ray's 300B coefficient block into LDS
// via gfx1250 GLOBAL_LOAD_ASYNC_TO_LDS_B32 (coalesced 128B per instruction):
//   SADDR = sh + rowByteOffset(ray r)   (scalar add, uniform per ray)
//   VADDR = lane*4                      (shared unsigned VGPR, never negative)
//   IOFFSET in {0,128,256}              (chunk select, applies to LDS+global)
//   LDS dst VGPR chained +300 per ray   (1 VALU per ray)
// SH-basis VALU work overlaps the DMA; s_wait_asynccnt; then a conflict-free
// LDS reduce (stride 75 dwords, gcd(75,64)=1 -> no bank conflicts).

#define WAVES_PER_BLOCK 4
#define THREADS (WAVES_PER_BLOCK * 32)
#define NC   25
#define NDW  75          // dwords per ray (25 coefs * 3 channels)
#define RESO 512

template<int IMM>
__device__ __forceinline__ void async_ld(unsigned lds, unsigned voff, const char* sa) {
    asm volatile("global_load_async_to_lds_b32 %0, %1, %2 offset:%c3"
                 :: "v"(lds), "v"(voff), "s"(sa), "n"(IMM) : "memory");
}

// Chunks 0 (dwords 0..31) and 1 (dwords 32..63) for all 32 rays of the wave.
template<int R>
__device__ __forceinline__ void gather_main(unsigned ldsOff, unsigned lane4,
                                            unsigned rowOff, const char* sh) {
    if constexpr (R < 32) {
        unsigned base = (unsigned)__builtin_amdgcn_readlane((int)rowOff, R);
        const char* sa = sh + base;              // uniform, always in-bounds
        async_ld<0  >(ldsOff, lane4, sa);
        async_ld<128>(ldsOff, lane4, sa);
        gather_main<R + 1>(ldsOff + 300u, lane4, rowOff, sh);
    }
}

// Chunk 2 (dwords 64..74, lanes 0..10) for all 32 rays of the wave.
template<int R>
__device__ __forceinline__ void gather_tail(unsigned ldsOff, unsigned lane4,
                                            unsigned rowOff, const char* sh) {
    if constexpr (R < 32) {
        unsigned base = (unsigned)__builtin_amdgcn_readlane((int)rowOff, R);
        const char* sa = sh + base;
        async_ld<256>(ldsOff, lane4, sa);
        gather_tail<R + 1>(ldsOff + 300u, lane4, rowOff, sh);
    }
}

__global__ __launch_bounds__(THREADS) void sh_embed_kernel(
    const float* __restrict__ yv,
    const float* __restrict__ xv,
    const float* __restrict__ rdir,
    const float* __restrict__ sh,     // (512,512,25,3) f32
    float* __restrict__ out,          // (B,3) f32
    int n)
{
    __shared__ float smem[WAVES_PER_BLOCK * 32 * NDW];   // 38400 B

    const int lane = threadIdx.x & 31;
    const int wave = threadIdx.x >> 5;
    const int ray  = blockIdx.x * THREADS + threadIdx.x;
    const int rayc = ray < n ? ray : (n - 1);            // clamp for loads

    // ---- texel lookup -> byte offset of this ray's 300B coef block ----
    float yf = __builtin_nontemporal_load(&yv[rayc]);
    float xf = __builtin_nontemporal_load(&xv[rayc]);
    int yi = (int)rintf(yf); yi = min(max(yi, 0), RESO - 1);
    int xi = (int)rintf(xf); xi = min(max(xi, 0), RESO - 1);
    unsigned rowOff = (unsigned)(yi * RESO + xi) * (NDW * 4u);

    const unsigned lane4   = (unsigned)lane * 4u;
    const unsigned ldsBase =
        (unsigned)(uintptr_t)(&smem[wave * 32 * NDW]) + lane4;

    // ---- cooperative async gather: ray r of this wave -> LDS r*300 ----
    gather_main<0>(ldsBase, lane4, rowOff, (const char*)sh);
    if (lane < 11) {
        gather_tail<0>(ldsBase, lane4, rowOff, (const char*)sh);
    }

    // ---- SH basis (overlaps the async DMA) ----
    float dx = __builtin_nontemporal_load(&rdir[rayc * 3 + 0]);
    float dy = __builtin_nontemporal_load(&rdir[rayc * 3 + 1]);
    float dz = __builtin_nontemporal_load(&rdir[rayc * 3 + 2]);

    float rn  = sqrtf(dx * dx + dy * dy + dz * dz) + 1e-14f;
    float ct  = fminf(fmaxf(dz / rn, -1.f), 1.f);
    float st2 = fmaxf(1.f - ct * ct, 1e-14f);
    float st  = sqrtf(st2);

    float rho2 = dx * dx + dy * dy;
    float c1, s1;
    if (rho2 > 0.f) { float ir = rsqrtf(rho2); c1 = dx * ir; s1 = dy * ir; }
    else            { c1 = 1.f; s1 = 0.f; }
    float c2 = c1 * c1 - s1 * s1, s2 = 2.f * c1 * s1;
    float c3 = c2 * c1 - s2 * s1, s3 = s2 * c1 + c2 * s1;
    float c4 = c2 * c2 - s2 * s2, s4 = 2.f * c2 * s2;

    float ct2 = ct * ct;
    float st3 = st2 * st;
    float st4 = st2 * st2;

    float b[NC];
    b[0]  =  0.28209479177f;
    b[1]  = -0.48860251190f * st * s1;
    b[2]  =  0.48860251190f * ct;
    b[3]  = -0.48860251190f * st * c1;
    b[4]  =  0.54627421529f * st2 * s2;
    b[5]  = -1.09254843059f * ct * st * s1;
    b[6]  =  0.31539156525f * (3.f * ct2 - 1.f);
    b[7]  = -1.09254843059f * ct * st * c1;
    b[8]  =  0.54627421529f * st2 * c2;
    b[9]  = -0.59004358993f * st3 * s3;
    b[10] =  1.44530572132f * ct * st2 * s2;
    b[11] = -0.45704579946f * (5.f * ct2 - 1.f) * st * s1;
    b[12] =  0.37317633259f * ct * (5.f * ct2 - 3.f);
    b[13] = -0.45704579946f * (5.f * ct2 - 1.f) * st * c1;
    b[14] =  1.44530572132f * ct * st2 * c2;
    b[15] = -0.59004358993f * st3 * c3;
    b[16] =  0.62583573544f * st4 * s4;
    b[17] = -1.77013076978f * ct * st3 * s3;
    b[18] =  0.47308734787f * (7.f * ct2 - 1.f) * st2 * s2;
    b[19] = -0.66904654355f * ct * (7.f * ct2 - 3.f) * st * s1;
    b[20] =  0.10578554691f * ((35.f * ct2 - 30.f) * ct2 + 3.f);
    b[21] = -0.66904654355f * ct * (7.f * ct2 - 3.f) * st * c1;
    b[22] =  0.47308734787f * (7.f * ct2 - 1.f) * st2 * c2;
    b[23] = -1.77013076978f * ct * st3 * c3;
    b[24] =  0.62583573544f * st4 * c4;

    // ---- wait for DMA, then conflict-free LDS reduce ----
    asm volatile("s_wait_asynccnt 0x0" ::: "memory");

    const float* my = &smem[wave * 32 * NDW + lane * NDW];
    float ar = 0.f, ag = 0.f, ab = 0.f;
#pragma unroll
    for (int k = 0; k < NC; ++k) {
        ar = fmaf(b[k], my[3 * k + 0], ar);
        ag = fmaf(b[k], my[3 * k + 1], ag);
        ab = fmaf(b[k], my[3 * k + 2], ab);
    }

    if (ray < n) {
        __builtin_nontemporal_store(fminf(fmaxf(ar, 0.f), 1.f), &out[ray * 3 + 0]);
        __builtin_nontemporal_store(fminf(fmaxf(ag, 0.f), 1.f), &out[ray * 3 + 1]);
        __builtin_nontemporal_store(fminf(fmaxf(ab, 0.f), 1.f), &out[ray * 3 + 2]);
    }
}

extern "C" void kernel_launch(void* const* d_in, const int* in_sizes, int n_in,
                              void* d_out, int out_size, void* d_ws, size_t ws_size,
                              hipStream_t stream) {
    const float* y   = (const float*)d_in[0];
    const float* x   = (const float*)d_in[1];
    const float* rd  = (const float*)d_in[2];
    const float* sh  = (const float*)d_in[3];
    float* out = (float*)d_out;
    const int n = in_sizes[0];

    const int grid = (n + THREADS - 1) / THREADS;
    sh_embed_kernel<<<grid, THREADS, 0, stream>>>(y, x, rd, sh, out, n);
}